// MaskPostProcessor_9045201125715
// MI455X (gfx1250) — compile-verified
//
#include <hip/hip_runtime.h>
#include <math.h>

// Problem constants (from reference): N=512 dets, C=81 classes, M=28 -> 784 px
#define NDET 512
#define NCLS 81
#define MM   784                       // 28*28
#define CH_STRIDE (NCLS * MM)          // per-detection stride in logits
#define PROBS_PER_SIDE (NDET * MM)     // 401408
#define OUT_KEEP_OFF (2 * PROBS_PER_SIDE)
#define NMS_THRESH 0.5f
#define EPS_F 1e-4f

typedef __attribute__((ext_vector_type(2))) float v2f;
typedef __attribute__((ext_vector_type(8))) float v8f;

// ---------------------------------------------------------------------------
// Pass 1: gather class channel, sigmoid, write probs output, row-sum (mask area)
// One block per (side, detection). Only touches the gathered channel: the
// reference's full-tensor sigmoid over all 81 classes is dead work.
// ---------------------------------------------------------------------------
__global__ __launch_bounds__(256) void gather_sigmoid_kernel(
    const float* __restrict__ left, const float* __restrict__ right,
    const int* __restrict__ labels, float* __restrict__ out_probs,
    float* __restrict__ u) {
  int b = blockIdx.x;              // 0..1023
  int side = b >> 9;
  int det = b & (NDET - 1);
  const float* src = (side == 0 ? left : right) +
                     (size_t)det * CH_STRIDE + (size_t)labels[det] * MM;
  float* dst = out_probs + (size_t)side * PROBS_PER_SIDE + (size_t)det * MM;

  __shared__ float red[256];
  float s = 0.f;
  for (int p = threadIdx.x; p < MM; p += 256) {
    float pr = 1.0f / (1.0f + expf(-src[p]));
    dst[p] = pr;
    s += pr;
  }
  red[threadIdx.x] = s;
  __syncthreads();
  for (int off = 128; off > 0; off >>= 1) {
    if (threadIdx.x < off) red[threadIdx.x] += red[threadIdx.x + off];
    __syncthreads();
  }
  if (threadIdx.x == 0) u[side * NDET + det] = red[0];
}

// ---------------------------------------------------------------------------
// Pass 2: stable descending argsort of 512 scores via O(N^2) rank (matches
// jnp.argsort(-scores) tie-breaking: equal scores -> lower index first).
// ---------------------------------------------------------------------------
__global__ __launch_bounds__(512) void rank_kernel(
    const float* __restrict__ scores, int* __restrict__ order) {
  __shared__ float s[NDET];
  int i = threadIdx.x;
  s[i] = scores[i];
  __syncthreads();
  float si = s[i];
  int r = 0;
  for (int j = 0; j < NDET; ++j) {
    float sj = s[j];
    r += (sj > si) || (sj == si && j < i);
  }
  order[r] = i;
}

// ---------------------------------------------------------------------------
// Pass 3: inter = P_sorted @ P_sorted^T via V_WMMA_F32_16X16X4_F32.
// One wave per 16x16 output tile; rows gathered through order[] so the sorted
// P matrix is never materialized. K = 784 = 196 x 4.
// A 16x4 fragment: lane L -> row (L&15), K pair 2*(L>>4)+{0,1}  (8B load)
// B 4x16 fragment: mirrored (same addressing with the N-tile base).
// EXEC is all-ones: exact grid, no divergence around the WMMA.
// ---------------------------------------------------------------------------
__global__ __launch_bounds__(256) void inter_wmma_kernel(
    const float* __restrict__ probs, const int* __restrict__ order,
    float* __restrict__ inter) {
  int wave = (blockIdx.x << 3) + (threadIdx.x >> 5);  // 0..2047
  int lane = threadIdx.x & 31;
  int side = wave >> 10;                              // 0 = left, 1 = right
  int t = wave & 1023;
  int tm = (t >> 5) << 4;                             // output row tile base
  int tn = (t & 31) << 4;                             // output col tile base
  int half = lane >> 4;
  int l15 = lane & 15;

  const float* base = probs + (size_t)side * PROBS_PER_SIDE;
  const float* rowA = base + (size_t)order[tm + l15] * MM + 2 * half;
  const float* rowB = base + (size_t)order[tn + l15] * MM + 2 * half;

  v8f c = {};
  for (int k = 0; k < MM; k += 4) {
    v2f a = *(const v2f*)(rowA + k);   // 8B-aligned: global_load_b64
    v2f b = *(const v2f*)(rowB + k);
    // 8 args: (neg_a, A, neg_b, B, c_mod, C, reuse_a, reuse_b)
    c = __builtin_amdgcn_wmma_f32_16x16x4_f32(
        false, a, false, b, (short)0, c, false, false);
  }

  float* dst = inter + (size_t)side * (NDET * NDET);
#pragma unroll
  for (int k = 0; k < 8; ++k) {
    int m = tm + k + 8 * half;                        // C layout: VGPR k row
    dst[(size_t)m * NDET + tn + l15] = c[k];
  }
}

// ---------------------------------------------------------------------------
// Pass 4: greedy mask-NMS in sorted order. One block per side, thread j owns
// sorted slot j. iou[i][j] >= 0.5  <=>  inter[i][j] >= 0.5*(u_sorted[i]+eps).
// Writes keep flag scattered back to original detection index.
// ---------------------------------------------------------------------------
__global__ __launch_bounds__(512) void nms_kernel(
    const float* __restrict__ inter, const float* __restrict__ u,
    const int* __restrict__ order, int* __restrict__ keep_side) {
  int side = blockIdx.x;
  int j = threadIdx.x;
  __shared__ float th[NDET];
  __shared__ int sup[NDET];
  int oj = order[j];
  th[j] = NMS_THRESH * (u[side * NDET + oj] + EPS_F);
  sup[j] = 0;
  __syncthreads();

  const float* row0 = inter + (size_t)side * (NDET * NDET);
  for (int i = 0; i < NDET; ++i) {
    int si = sup[i];                       // only j>i writes this iter: safe
    float v = row0[(size_t)i * NDET + j];  // coalesced row sweep
    if (!si && j > i && v >= th[i]) sup[j] = 1;
    __syncthreads();
  }
  keep_side[side * NDET + oj] = !sup[j];
}

// ---------------------------------------------------------------------------
// Pass 5: keep = valid & left_keep & right_keep  (valid: mask mass > 0)
// ---------------------------------------------------------------------------
__global__ __launch_bounds__(256) void finalize_kernel(
    const float* __restrict__ u, const int* __restrict__ keep_side,
    float* __restrict__ out_keep) {
  int i = blockIdx.x * 256 + threadIdx.x;
  if (i >= NDET) return;
  bool valid = (u[i] > 0.f) && (u[NDET + i] > 0.f);
  bool k = valid && keep_side[i] && keep_side[NDET + i];
  out_keep[i] = k ? 1.0f : 0.0f;
}

// ---------------------------------------------------------------------------
extern "C" void kernel_launch(void* const* d_in, const int* in_sizes, int n_in,
                              void* d_out, int out_size, void* d_ws,
                              size_t ws_size, hipStream_t stream) {
  const float* left   = (const float*)d_in[0];
  const float* right  = (const float*)d_in[1];
  const float* scores = (const float*)d_in[2];
  const int*   labels = (const int*)d_in[3];
  float* out = (float*)d_out;  // [2*512*784 probs][512 keep-as-float]

  // Workspace layout (bytes): u 4096 | order 2048 | keep_side 4096 | inter 2MB
  char* ws = (char*)d_ws;
  float* u        = (float*)(ws + 0);
  int*   order    = (int*)(ws + 4096);
  int*   keepSide = (int*)(ws + 6144);
  float* inter    = (float*)(ws + 10240);

  gather_sigmoid_kernel<<<1024, 256, 0, stream>>>(left, right, labels, out, u);
  rank_kernel<<<1, 512, 0, stream>>>(scores, order);
  inter_wmma_kernel<<<256, 256, 0, stream>>>(out, order, inter);
  nms_kernel<<<2, 512, 0, stream>>>(inter, u, order, keepSide);
  finalize_kernel<<<2, 256, 0, stream>>>(u, keepSide, out + OUT_KEEP_OFF);
}